// CLAHE_59347858096781
// MI455X (gfx1250) — compile-verified
//
#include <hip/hip_runtime.h>

#define HI 4096
#define WI 4096
#define NTILES 8
#define TH 512
#define NBINS 256
#define CLIPV 40960u            // round(40.0 * 512*512 / 256)
#define TILE_AREA (512*512)

#define USE_TDM 1               // flip to 0 to fall back to per-lane async-LDS loads

typedef __attribute__((ext_vector_type(2))) float v2f;
typedef __attribute__((ext_vector_type(8))) float v8f;
typedef __attribute__((ext_vector_type(4))) unsigned int v4u;
typedef __attribute__((ext_vector_type(8))) int v8i;
typedef __attribute__((ext_vector_type(4))) int v4i;

// ---------------- K0: zero the 64x256 u32 histogram in workspace ----------------
__global__ void clahe_zero(unsigned* __restrict__ hist) {
    int i = blockIdx.x * blockDim.x + threadIdx.x;
    hist[i] = 0u;
}

// ---------------- K1: per-tile histograms (LDS-privatized, ds_add) ----------------
// 2048 blocks: 32 slabs (512x16) per 512x512 tile. 256 thr/block, 8 float4 each.
__global__ __launch_bounds__(256) void clahe_hist(const float* __restrict__ x,
                                                  unsigned* __restrict__ ghist) {
    __shared__ unsigned hloc[NBINS];
    const int tid = threadIdx.x;
    hloc[tid] = 0u;
    __syncthreads();

    const int bid   = blockIdx.x;
    const int tile  = bid >> 5;          // 0..63
    const int slab  = bid & 31;          // 0..31
    const int tileY = tile >> 3, tileX = tile & 7;
    const int rowstart  = tileY * TH + slab * 16;
    const int colstart4 = tileX * (TH / 4);
    const float4* __restrict__ xin = (const float4*)x;

    #pragma unroll
    for (int k = 0; k < 8; ++k) {
        int f    = tid + k * 256;        // 0..2047 float4 within slab
        int row  = f >> 7;               // 128 float4 per slab row
        int col4 = f & 127;
        float4 p = xin[(size_t)(rowstart + row) * (WI / 4) + colstart4 + col4];
        int v0 = min(max((int)floorf(p.x * 256.0f), 0), 255);
        int v1 = min(max((int)floorf(p.y * 256.0f), 0), 255);
        int v2 = min(max((int)floorf(p.z * 256.0f), 0), 255);
        int v3 = min(max((int)floorf(p.w * 256.0f), 0), 255);
        atomicAdd(&hloc[v0], 1u);
        atomicAdd(&hloc[v1], 1u);
        atomicAdd(&hloc[v2], 1u);
        atomicAdd(&hloc[v3], 1u);
    }
    __syncthreads();
    atomicAdd(&ghist[tile * NBINS + tid], hloc[tid]);
}

// ---------------- K2: clip + OpenCV excess redistribution -> float hist ----------------
__global__ __launch_bounds__(256) void clahe_clip(const unsigned* __restrict__ ghist,
                                                  float* __restrict__ histf) {
    __shared__ unsigned sred[NBINS];
    const int tile = blockIdx.x;
    const int i    = threadIdx.x;
    unsigned h       = ghist[tile * NBINS + i];
    unsigned clipped = (h < CLIPV) ? h : CLIPV;
    sred[i] = h - clipped;
    __syncthreads();
    for (int s = 128; s > 0; s >>= 1) {
        if (i < s) sred[i] += sred[i + s];
        __syncthreads();
    }
    unsigned excess   = sred[0];
    unsigned h2       = clipped + excess / NBINS;
    unsigned residual = excess & (NBINS - 1);
    unsigned rdiv     = (residual > 0u) ? residual : 1u;
    unsigned step     = 256u / rdiv;
    if (step < 1u) step = 1u;
    if (((unsigned)i % step) == 0u && ((unsigned)i / step) < residual) h2 += 1u;
    histf[tile * NBINS + i] = (float)h2;   // exact: h2 <= 262144 < 2^24
}

// ---------------- K3: CDF = H(64x256) x UpperTri(256x256) via V_WMMA_F32_16X16X4_F32 ----
// 64 blocks x 32 lanes: block (mt,nt) computes 16x16 tile of CDF, then LUT = rint(cdf*scale).
__global__ __launch_bounds__(32) void clahe_cdf_wmma(const float* __restrict__ histf,
                                                     float* __restrict__ lut) {
    const int lane  = threadIdx.x;           // 0..31, wave32, EXEC all-ones
    const int mt    = blockIdx.x >> 4;       // 0..3
    const int nt    = blockIdx.x & 15;       // 0..15
    const int m     = mt * 16 + (lane & 15); // A-matrix row (tile id)
    const int n     = nt * 16 + (lane & 15); // B/D column (bin)
    const int khalf = (lane >> 4) * 2;       // lanes 0-15: K=0,1 ; lanes 16-31: K=2,3

    v8f c = {};
    for (int k4 = 0; k4 < NBINS; k4 += 4) {
        v2f a, b;
        a.x = histf[m * NBINS + k4 + khalf];
        a.y = histf[m * NBINS + k4 + khalf + 1];
        b.x = ((k4 + khalf)     <= n) ? 1.0f : 0.0f;   // upper-triangular ones
        b.y = ((k4 + khalf + 1) <= n) ? 1.0f : 0.0f;
        c = __builtin_amdgcn_wmma_f32_16x16x4_f32(false, a, false, b,
                                                  (short)0, c, false, false);
    }

    const float scale = 255.0f / (float)TILE_AREA;
    #pragma unroll
    for (int r = 0; r < 8; ++r) {
        int mrel   = r + ((lane >> 4) << 3);       // C layout: lanes>=16 hold M=r+8
        int tile_m = mt * 16 + mrel;
        int bin    = nt * 16 + (lane & 15);
        float val  = rintf(c[r] * scale);
        val = fminf(fmaxf(val, 0.0f), 255.0f);
        lut[tile_m * NBINS + bin] = val;
    }
}

// ---------------- K4: apply — stage 64KB LUT into LDS via TDM, bilinear blend ----------
// 512 blocks x 8 rows each: staging traffic amortized 8x vs one-row blocks.
__global__ __launch_bounds__(256) void clahe_apply(const float* __restrict__ x,
                                                   const float* __restrict__ lutg,
                                                   float* __restrict__ out) {
    __shared__ float slut[NTILES * NTILES * NBINS];   // 16384 floats = 64 KB
    const int tid = threadIdx.x;
    unsigned ldsbase = (unsigned)(size_t)(&slut[0]);  // low 32 bits of flat shared ptr = LDS offset

#if USE_TDM
    // Tensor Data Mover: one DMA op (wave 0) copies the whole 64KB LUT to LDS.
    // D# per CDNA5 ISA: 2D tensor, data_size=4B,
    // tensor 4096x4, tile 4096x4, stride0=4096 elements, contiguous LDS dest.
    if (tid < 32) {
        unsigned long long ga = (unsigned long long)(size_t)lutg;
        v4u g0;
        g0.x = 1u;                                        // count=1, is_restore=0, gather off
        g0.y = ldsbase;                                   // lds_addr (bytes)
        g0.z = (unsigned)(ga & 0xFFFFFFFFull);            // global_addr[31:0]
        g0.w = (unsigned)((ga >> 32) & 0x1FFFFFFull)      // global_addr[56:32]
             | (2u << 30);                                // type=2 ("image")
        v8i g1;
        g1[0] = (int)(2u << 16);                          // wg_mask=0, data_size=2 (4B)
        g1[1] = (int)((4096u & 0xFFFFu) << 16);           // tensor_dim0[15:0]=4096
        g1[2] = (int)(4u << 16);                          // tensor_dim0 hi=0 | tensor_dim1=4
        g1[3] = (int)(4096u << 16);                       // tensor_dim1 hi=0 | tile_dim0=4096
        g1[4] = 4;                                        // tile_dim1=4, tile_dim2=0
        g1[5] = 4096;                                     // tensor_dim0_stride[31:0]
        g1[6] = 0;                                        // stride0 hi | stride1 lo
        g1[7] = 0;
        v4i g2 = {0, 0, 0, 0};
        v4i g3 = {0, 0, 0, 0};
#if defined(__clang_major__) && (__clang_major__ >= 23)
        v8i g4 = {0, 0, 0, 0, 0, 0, 0, 0};
        __builtin_amdgcn_tensor_load_to_lds(g0, g1, g2, g3, g4, 0);
#else
        __builtin_amdgcn_tensor_load_to_lds(g0, g1, g2, g3, 0);
#endif
        __builtin_amdgcn_s_wait_tensorcnt(0);
    }
#else
    // Fallback: per-lane async global->LDS copy (16 x b128 per thread = 64 KB/block)
    #pragma unroll
    for (int i = 0; i < 16; ++i) {
        unsigned off  = ((unsigned)(i * 256 + tid)) * 16u;
        unsigned ldst = ldsbase + off;
        const char* g = (const char*)lutg + off;
        asm volatile("global_load_async_to_lds_b128 %0, %1, off"
                     :: "v"(ldst), "v"(g) : "memory");
    }
    asm volatile("s_wait_asynccnt 0" ::: "memory");
#endif
    __syncthreads();

    #pragma unroll
    for (int r = 0; r < 8; ++r) {
        const int y = blockIdx.x * 8 + r;
        float tyf  = (float)y * (1.0f / TH) - 0.5f;
        float tyfl = floorf(tyf);
        float ya   = tyf - tyfl;
        int ty1 = (int)tyfl;
        int ty2 = min(max(ty1 + 1, 0), NTILES - 1);
        ty1     = min(max(ty1, 0), NTILES - 1);
        const float* __restrict__ rowA = &slut[ty1 * NTILES * NBINS];
        const float* __restrict__ rowB = &slut[ty2 * NTILES * NBINS];

        const float4* __restrict__ xin  = (const float4*)(x   + (size_t)y * WI);
        float4* __restrict__       outp = (float4*)      (out + (size_t)y * WI);

        #pragma unroll
        for (int j = 0; j < 4; ++j) {
            int f = tid + j * 256;                 // 0..1023 float4 in row
            float4 p = xin[f];
            float4 o;
            const float* pe = (const float*)&p;
            float*       oe = (float*)&o;
            #pragma unroll
            for (int e = 0; e < 4; ++e) {
                int xc = f * 4 + e;
                int v  = min(max((int)floorf(pe[e] * 256.0f), 0), 255);
                float txf  = (float)xc * (1.0f / TH) - 0.5f;
                float txfl = floorf(txf);
                float xa   = txf - txfl;
                int tx1 = (int)txfl;
                int tx2 = min(max(tx1 + 1, 0), NTILES - 1);
                tx1     = min(max(tx1, 0), NTILES - 1);
                float l11 = rowA[tx1 * NBINS + v];
                float l12 = rowA[tx2 * NBINS + v];
                float l21 = rowB[tx1 * NBINS + v];
                float l22 = rowB[tx2 * NBINS + v];
                // match reference FP expression ordering exactly
                float top = l11 * (1.0f - xa) + l12 * xa;
                float bot = l21 * (1.0f - xa) + l22 * xa;
                oe[e] = rintf(top * (1.0f - ya) + bot * ya);
            }
            outp[f] = o;
        }
    }
}

extern "C" void kernel_launch(void* const* d_in, const int* in_sizes, int n_in,
                              void* d_out, int out_size, void* d_ws, size_t ws_size,
                              hipStream_t stream) {
    (void)in_sizes; (void)n_in; (void)out_size; (void)ws_size;
    const float* x = (const float*)d_in[0];
    float* out = (float*)d_out;

    unsigned* hist  = (unsigned*)d_ws;                       // 64*256 u32 = 64 KB
    float*    histf = (float*)((char*)d_ws + (64 * 1024));   // 64*256 f32 = 64 KB
    float*    lut   = (float*)((char*)d_ws + (128 * 1024));  // 64*256 f32 = 64 KB

    clahe_zero    <<<64,   256, 0, stream>>>(hist);
    clahe_hist    <<<2048, 256, 0, stream>>>(x, hist);
    clahe_clip    <<<64,   256, 0, stream>>>(hist, histf);
    clahe_cdf_wmma<<<64,    32, 0, stream>>>(histf, lut);
    clahe_apply   <<<512,  256, 0, stream>>>(x, lut, out);
}